// X_layer_1924145349076
// MI455X (gfx1250) — compile-verified
//
#include <hip/hip_runtime.h>

typedef __attribute__((ext_vector_type(2))) float v2f;
typedef __attribute__((ext_vector_type(8))) float v8f;

#define KB 512
#define NN 512
#define DD 16
#define MM 64
#define MRF 16
#define NRF 64

// D = A(16x4) * B(4x16) + C, f32 WMMA (CDNA5, wave32)
#define WMMA_F32(a, b, c) \
  __builtin_amdgcn_wmma_f32_16x16x4_f32(false, (a), false, (b), (short)0, (c), false, false)

static __device__ __forceinline__ int lane() { return (int)(threadIdx.x & 31u); }
// Wave index as a *scalar* (SGPR): keeps wave-keyed branches/loops in SALU so
// EXEC stays all-ones through every WMMA (ISA 7.12 requirement), and lets tile
// base addresses fold into scalar offsets.
static __device__ __forceinline__ int waveS() {
  return __builtin_amdgcn_readfirstlane((int)(threadIdx.x >> 5));
}

// A-fragment (16x4) of row-major src, tile origin (r0,c0), leading dim ld.
// lane L: row = r0 + (L&15); k-cols = c0 + ((L>=16)?2:0) + {0,1}
static __device__ __forceinline__ v2f ldA(const float* s, int ld, int r0, int c0) {
  int l = lane();
  int r = r0 + (l & 15);
  int c = c0 + ((l >> 4) << 1);
  v2f a;
  a.x = s[(size_t)r * ld + c];
  a.y = s[(size_t)r * ld + c + 1];
  return a;
}

// B-fragment (4x16) of row-major src, tile origin (r0,c0).
// lane L: col = c0 + (L&15); k-rows = r0 + ((L>=16)?2:0) + {0,1}
// NOTE: ldB(src,...) also yields the A-fragment of src^T (and ldA yields the
// B-fragment of src^T) — used for H^H and temp^H without materializing transposes.
static __device__ __forceinline__ v2f ldB(const float* s, int ld, int r0, int c0) {
  int l = lane();
  int c = c0 + (l & 15);
  int r = r0 + ((l >> 4) << 1);
  v2f b;
  b.x = s[(size_t)r * ld + c];
  b.y = s[(size_t)(r + 1) * ld + c];
  return b;
}

static __device__ __forceinline__ v2f vneg(v2f x) {
  v2f y; y.x = -x.x; y.y = -x.y; return y;
}

// C-fragment (16x16, 8 VGPRs): VGPR j -> row r0 + j + ((L>=16)?8:0), col c0 + (L&15)
static __device__ __forceinline__ void stC(float* d, int ld, int r0, int c0, v8f c) {
  int l = lane();
  int cc = c0 + (l & 15);
  int rb = r0 + ((l >> 4) << 3);
#pragma unroll
  for (int j = 0; j < 8; ++j) d[(size_t)(rb + j) * ld + cc] = c[j];
}

static __device__ __forceinline__ v8f ldC(const float* s, int ld, int r0, int c0) {
  int l = lane();
  int cc = c0 + (l & 15);
  int rb = r0 + ((l >> 4) << 3);
  v8f c;
#pragma unroll
  for (int j = 0; j < 8; ++j) c[j] = s[(size_t)(rb + j) * ld + cc];
  return c;
}

// ---------------------------------------------------------------------------
// Kernel 1: G = U_RF * U_BB (complex, 64x16, in LDS), then temp = H^H * G
// (512x16 complex per k). One block per k, 8 waves.
// temp_r = Hr^T Gr + Hi^T Gi ; temp_i = Hr^T Gi - Hi^T Gr
// ---------------------------------------------------------------------------
__global__ __launch_bounds__(256) void k_temp(const float* __restrict__ H,
                                              const float* __restrict__ U_RF,
                                              const float* __restrict__ U_BB,
                                              float* __restrict__ temp) {
  __shared__ float G[2][MM][DD];  // 8 KB of the 320 KB/WGP LDS
  const int k = blockIdx.x;
  const int w = waveS();  // scalar

  const float* Ur = U_RF + (size_t)k * 2 * MM * MRF;
  const float* Ui = Ur + MM * MRF;
  const float* Br = U_BB + (size_t)k * 2 * MRF * DD;
  const float* Bi = Br + MRF * DD;

  {  // waves 0-3: Gr tiles, waves 4-7: Gi tiles (scalar branch, EXEC untouched)
    const int m0 = (w & 3) * 16;
    v8f c = {};
    if (w < 4) {  // Gr = Ur*UBr - Ui*UBi
#pragma unroll
      for (int s = 0; s < 4; ++s) {
        const int d0 = 4 * s;
        v2f aR = ldA(Ur, MRF, m0, d0);
        v2f aI = ldA(Ui, MRF, m0, d0);
        v2f bR = ldB(Br, DD, d0, 0);
        v2f bI = ldB(Bi, DD, d0, 0);
        c = WMMA_F32(aR, bR, c);
        c = WMMA_F32(aI, vneg(bI), c);
      }
      stC(&G[0][0][0], DD, m0, 0, c);
    } else {      // Gi = Ur*UBi + Ui*UBr
#pragma unroll
      for (int s = 0; s < 4; ++s) {
        const int d0 = 4 * s;
        v2f aR = ldA(Ur, MRF, m0, d0);
        v2f aI = ldA(Ui, MRF, m0, d0);
        v2f bR = ldB(Br, DD, d0, 0);
        v2f bI = ldB(Bi, DD, d0, 0);
        c = WMMA_F32(aR, bI, c);
        c = WMMA_F32(aI, bR, c);
      }
      stC(&G[1][0][0], DD, m0, 0, c);
    }
  }
  __syncthreads();

  const float* Hr = H + (size_t)k * 2 * MM * NN;
  const float* Hi = Hr + MM * NN;
  float* Tr = temp + (size_t)k * 2 * NN * DD;
  float* Ti = Tr + NN * DD;

  for (int t = w; t < NN / 16; t += 8) {  // scalar loop bounds
    const int n0 = t * 16;
    v8f cr = {}, ci = {};
#pragma unroll
    for (int s = 0; s < 16; ++s) {
      const int m0 = 4 * s;
      v2f aR = ldB(Hr, NN, m0, n0);  // A-fragment of Hr^T
      v2f aI = ldB(Hi, NN, m0, n0);  // A-fragment of Hi^T
      v2f bR = ldB(&G[0][0][0], DD, m0, 0);
      v2f bI = ldB(&G[1][0][0], DD, m0, 0);
      cr = WMMA_F32(aR, bR, cr);
      cr = WMMA_F32(aI, bI, cr);
      ci = WMMA_F32(aR, bI, ci);
      ci = WMMA_F32(aI, vneg(bR), ci);
    }
    stC(Tr, DD, n0, 0, cr);
    stC(Ti, DD, n0, 0, ci);
  }
}

// ---------------------------------------------------------------------------
// Kernel 2: TW = temp * W ; B_rho = TW + 0.5*(V_RF*V_BB/rho - Y)
// ---------------------------------------------------------------------------
__global__ __launch_bounds__(256) void k_tw_brho(const float* __restrict__ temp,
                                                 const float* __restrict__ W,
                                                 const float* __restrict__ V_RF,
                                                 const float* __restrict__ V_BB,
                                                 const float* __restrict__ Y,
                                                 const float* __restrict__ rhoX_p,
                                                 float* __restrict__ TW,
                                                 float* __restrict__ Brho) {
  const int k = blockIdx.x;
  const float invRho = 1.0f / rhoX_p[0];

  const float* tr = temp + (size_t)k * 2 * NN * DD;
  const float* ti = tr + NN * DD;
  const float* Wr = W + (size_t)k * 2 * DD * DD;
  const float* Wi = Wr + DD * DD;
  const float* Vr = V_RF;             // (512,64) shared across k
  const float* Vi = V_RF + NN * NRF;
  const float* VBr = V_BB + (size_t)k * 2 * NRF * DD;
  const float* VBi = VBr + NRF * DD;
  const float* Yr = Y + (size_t)k * 2 * NN * DD;
  const float* Yi = Yr + NN * DD;
  float* twr = TW + (size_t)k * 2 * NN * DD;
  float* twi = twr + NN * DD;
  float* br_ = Brho + (size_t)k * 2 * NN * DD;
  float* bi_ = br_ + NN * DD;

  for (int t = waveS(); t < NN / 16; t += 8) {
    const int n0 = t * 16;
    v8f twR = {}, twI = {};
#pragma unroll
    for (int s = 0; s < 4; ++s) {  // TW = temp*W (inner 16)
      const int d0 = 4 * s;
      v2f aR = ldA(tr, DD, n0, d0);
      v2f aI = ldA(ti, DD, n0, d0);
      v2f bR = ldB(Wr, DD, d0, 0);
      v2f bI = ldB(Wi, DD, d0, 0);
      twR = WMMA_F32(aR, bR, twR);
      twR = WMMA_F32(aI, vneg(bI), twR);
      twI = WMMA_F32(aR, bI, twI);
      twI = WMMA_F32(aI, bR, twI);
    }
    v8f vpR = {}, vpI = {};
#pragma unroll
    for (int s = 0; s < 16; ++s) {  // Vp = V_RF*V_BB (inner 64)
      const int m0 = 4 * s;
      v2f aR = ldA(Vr, NRF, n0, m0);
      v2f aI = ldA(Vi, NRF, n0, m0);
      v2f bR = ldB(VBr, DD, m0, 0);
      v2f bI = ldB(VBi, DD, m0, 0);
      vpR = WMMA_F32(aR, bR, vpR);
      vpR = WMMA_F32(aI, vneg(bI), vpR);
      vpI = WMMA_F32(aR, bI, vpI);
      vpI = WMMA_F32(aI, bR, vpI);
    }
    v8f yR = ldC(Yr, DD, n0, 0);
    v8f yI = ldC(Yi, DD, n0, 0);
    v8f bRo, bIo;
#pragma unroll
    for (int j = 0; j < 8; ++j) {
      bRo[j] = twR[j] + 0.5f * (invRho * vpR[j] - yR[j]);
      bIo[j] = twI[j] + 0.5f * (invRho * vpI[j] - yI[j]);
    }
    stC(twr, DD, n0, 0, twR);
    stC(twi, DD, n0, 0, twI);
    stC(br_, DD, n0, 0, bRo);
    stC(bi_, DD, n0, 0, bIo);
  }
}

// ---------------------------------------------------------------------------
// Kernel 3: A_rho = sum_k TW * temp^H, + ridge (1/(2 rho)+mu) I, written
// directly into the real 1024x1024 block matrix Mgj = [[Ar, -Ai],[Ai, Ar]].
// Grid = 64 blocks (8x8 tiles of 64x64); inner dim K*D = 8192.
// temp+TW together are 67 MB -> resident in the 192 MB L2, so fragments
// stream straight from L2 (prefetch hints on the k loop).
// ---------------------------------------------------------------------------
__global__ __launch_bounds__(256) void k_arho(const float* __restrict__ TW,
                                              const float* __restrict__ temp,
                                              const float* __restrict__ rhoX_p,
                                              const float* __restrict__ mu_p,
                                              float* __restrict__ Mgj) {
  const int mb = (blockIdx.x & 7) * 64;
  const int nb = (blockIdx.x >> 3) * 64;
  const int w = waveS();  // scalar
  const int m0 = mb + (w & 3) * 16;
  const int n0a = nb + ((w >> 2) * 2) * 16;
  const int n0b = n0a + 16;

  v8f cr0 = {}, ci0 = {}, cr1 = {}, ci1 = {};
  for (int k = 0; k < KB; ++k) {
    const float* twr = TW + (size_t)k * 2 * NN * DD;
    const float* twi = twr + NN * DD;
    const float* tr = temp + (size_t)k * 2 * NN * DD;
    const float* ti = tr + NN * DD;
    if (k + 1 < KB) {  // -> global_prefetch_b8 into the cache hierarchy
      __builtin_prefetch(TW + (size_t)(k + 1) * 2 * NN * DD + (size_t)m0 * DD, 0, 1);
      __builtin_prefetch(temp + (size_t)(k + 1) * 2 * NN * DD + (size_t)n0a * DD, 0, 1);
    }
#pragma unroll
    for (int s = 0; s < 4; ++s) {
      const int d0 = 4 * s;
      v2f aR = ldA(twr, DD, m0, d0);
      v2f aI = ldA(twi, DD, m0, d0);
      {
        v2f bR = ldA(tr, DD, n0a, d0);  // B-fragment of temp^T
        v2f bI = ldA(ti, DD, n0a, d0);
        cr0 = WMMA_F32(aR, bR, cr0);    // Cr += TWr*tr^T + TWi*ti^T
        cr0 = WMMA_F32(aI, bI, cr0);
        ci0 = WMMA_F32(aI, bR, ci0);    // Ci += TWi*tr^T - TWr*ti^T
        ci0 = WMMA_F32(aR, vneg(bI), ci0);
      }
      {
        v2f bR = ldA(tr, DD, n0b, d0);
        v2f bI = ldA(ti, DD, n0b, d0);
        cr1 = WMMA_F32(aR, bR, cr1);
        cr1 = WMMA_F32(aI, bI, cr1);
        ci1 = WMMA_F32(aI, bR, ci1);
        ci1 = WMMA_F32(aR, vneg(bI), ci1);
      }
    }
  }

  const float cd = 0.5f / rhoX_p[0] + mu_p[0];
  const int l = lane();
  auto emit = [&](int n0, v8f cr, v8f ci) {
    const int cc = n0 + (l & 15);
    const int rb = m0 + ((l >> 4) << 3);
#pragma unroll
    for (int j = 0; j < 8; ++j) {
      const int rr = rb + j;
      const float re = cr[j] + ((rr == cc) ? cd : 0.0f);
      const float im = ci[j];
      Mgj[(size_t)rr * 1024 + cc] = re;
      Mgj[(size_t)rr * 1024 + (cc + 512)] = -im;
      Mgj[(size_t)(rr + 512) * 1024 + cc] = im;
      Mgj[(size_t)(rr + 512) * 1024 + (cc + 512)] = re;
    }
  };
  emit(n0a, cr0, ci0);
  emit(n0b, cr1, ci1);
}

// ---------------------------------------------------------------------------
// Blocked in-place Gauss-Jordan inversion of the 1024x1024 matrix, 16x16
// pivot blocks (64 steps). The O(n^3) work (rank-16 updates) is WMMA.
// ---------------------------------------------------------------------------
__global__ __launch_bounds__(256) void gj_diag(float* __restrict__ M, int p) {
  __shared__ float aug[16][33];  // [16][0:16]=A, [16][16:32]=I (padded)
  const int tid = threadIdx.x;
  for (int e = tid; e < 512; e += 256) {
    const int r = e >> 5, c = e & 31;
    float v;
    if (c < 16) v = M[(size_t)(p * 16 + r) * 1024 + (p * 16 + c)];
    else        v = (c - 16 == r) ? 1.0f : 0.0f;
    aug[r][c] = v;
  }
  __syncthreads();
  const int r0 = tid >> 5, c0 = tid & 31;
  const int r1 = (tid + 256) >> 5, c1 = (tid + 256) & 31;
  for (int it = 0; it < 16; ++it) {
    const float pinv = 1.0f / aug[it][it];
    __syncthreads();
    if (tid < 32) aug[it][tid] *= pinv;
    __syncthreads();
    const float f0 = aug[r0][it];
    const float f1 = aug[r1][it];
    __syncthreads();
    if (r0 != it) aug[r0][c0] -= f0 * aug[it][c0];
    if (r1 != it) aug[r1][c1] -= f1 * aug[it][c1];
    __syncthreads();
  }
  {
    const int r = tid >> 4, c = tid & 15;
    if (tid < 256)
      M[(size_t)(p * 16 + r) * 1024 + (p * 16 + c)] = aug[r][c + 16];  // Pinv in place
  }
}

// row panel: M[p, j] = Pinv * M[p, j]  (j != p). One wave per 16x16 tile.
__global__ __launch_bounds__(32) void gj_rowpanel(float* __restrict__ M, int p) {
  const int j = blockIdx.x;
  if (j == p) return;
  v8f c = {};
#pragma unroll
  for (int s = 0; s < 4; ++s) {
    v2f a = ldA(M, 1024, p * 16, p * 16 + 4 * s);       // Pinv
    v2f b = ldB(M, 1024, p * 16 + 4 * s, j * 16);
    c = WMMA_F32(a, b, c);
  }
  stC(M, 1024, p * 16, j * 16, c);
}

// trailing update: M[i, j] -= M[i, p] * M[p, j]  (i,j != p), then
// M[i, p] = -M[i, p] * Pinv. One block per row-panel i, 8 waves over j.
__global__ __launch_bounds__(256) void gj_update(float* __restrict__ M, int p) {
  const int i = blockIdx.x;
  if (i == p) return;
  const int w = waveS();  // scalar
  v2f aLn[4];
#pragma unroll
  for (int s = 0; s < 4; ++s)  // -L fragments (f32 WMMA has no A-negate)
    aLn[s] = vneg(ldA(M, 1024, i * 16, p * 16 + 4 * s));
  __syncthreads();  // everyone has read M[i,p] before wave 0 overwrites it

  for (int j = w; j < 64; j += 8) {  // scalar loop, scalar skip of j==p
    if (j == p) continue;
    v8f c = ldC(M, 1024, i * 16, j * 16);
#pragma unroll
    for (int s = 0; s < 4; ++s) {
      v2f b = ldB(M, 1024, p * 16 + 4 * s, j * 16);
      c = WMMA_F32(aLn[s], b, c);  // C += (-L) * R
    }
    stC(M, 1024, i * 16, j * 16, c);
  }
  if (w == 0) {  // column panel: M[i,p] = (-L) * Pinv  (scalar branch)
    v8f c = {};
#pragma unroll
    for (int s = 0; s < 4; ++s) {
      v2f b = ldB(M, 1024, p * 16 + 4 * s, p * 16);
      c = WMMA_F32(aLn[s], b, c);
    }
    stC(M, 1024, i * 16, p * 16, c);
  }
}

// ---------------------------------------------------------------------------
// Kernel 4: X = A_inv * B_rho + params4 (per k), A_inv from Mgj block form:
// Ar = Mgj[0:512,0:512], Ai = Mgj[512:1024,0:512]
// ---------------------------------------------------------------------------
__global__ __launch_bounds__(256) void k_x(const float* __restrict__ Minv,
                                           const float* __restrict__ Brho,
                                           const float* __restrict__ P4,
                                           float* __restrict__ X) {
  const int k = blockIdx.y;
  const int n0 = blockIdx.x * 128 + waveS() * 16;  // scalar row base
  const float* Ar = Minv;
  const float* Ai = Minv + (size_t)512 * 1024;
  const float* br = Brho + (size_t)k * 2 * NN * DD;
  const float* bi = br + NN * DD;

  v8f xr = {}, xi = {};
#pragma unroll 4
  for (int s = 0; s < 128; ++s) {
    const int m0 = 4 * s;
    v2f aR = ldA(Ar, 1024, n0, m0);
    v2f aI = ldA(Ai, 1024, n0, m0);
    v2f bR = ldB(br, DD, m0, 0);
    v2f bI = ldB(bi, DD, m0, 0);
    xr = WMMA_F32(aR, bR, xr);
    xr = WMMA_F32(aI, vneg(bI), xr);
    xi = WMMA_F32(aR, bI, xi);
    xi = WMMA_F32(aI, bR, xi);
  }
  const float* pr = P4 + (size_t)k * 2 * NN * DD;
  const float* pi = pr + NN * DD;
  v8f cr = ldC(pr, DD, n0, 0);
  v8f ci = ldC(pi, DD, n0, 0);
#pragma unroll
  for (int j = 0; j < 8; ++j) { xr[j] += cr[j]; xi[j] += ci[j]; }
  float* Xr = X + (size_t)k * 2 * NN * DD;
  float* Xi = Xr + NN * DD;
  stC(Xr, DD, n0, 0, xr);
  stC(Xi, DD, n0, 0, xi);
}

// ---------------------------------------------------------------------------
extern "C" void kernel_launch(void* const* d_in, const int* in_sizes, int n_in,
                              void* d_out, int out_size, void* d_ws, size_t ws_size,
                              hipStream_t stream) {
  const float* H    = (const float*)d_in[0];
  const float* U_RF = (const float*)d_in[1];
  const float* U_BB = (const float*)d_in[2];
  const float* W    = (const float*)d_in[3];
  const float* V_RF = (const float*)d_in[4];
  const float* V_BB = (const float*)d_in[5];
  const float* Y    = (const float*)d_in[6];
  // d_in[7] = rho (unused by the reference math)
  const float* rhoX = (const float*)d_in[8];
  const float* mu   = (const float*)d_in[9];
  const float* P4   = (const float*)d_in[10];

  float* ws = (float*)d_ws;
  const size_t PER = (size_t)2 * NN * DD;  // 16384 floats per k per array
  float* temp = ws;                        // (K,2,N,D)  33.5 MB
  float* TW   = ws + (size_t)KB * PER;     // (K,2,N,D)  33.5 MB
  float* Brho = ws + (size_t)2 * KB * PER; // (K,2,N,D)  33.5 MB
  float* Mgj  = ws + (size_t)3 * KB * PER; // 1024x1024  4 MB

  k_temp<<<KB, 256, 0, stream>>>(H, U_RF, U_BB, temp);
  k_tw_brho<<<KB, 256, 0, stream>>>(temp, W, V_RF, V_BB, Y, rhoX, TW, Brho);
  k_arho<<<64, 256, 0, stream>>>(TW, temp, rhoX, mu, Mgj);
  for (int p = 0; p < 64; ++p) {
    gj_diag<<<1, 256, 0, stream>>>(Mgj, p);
    gj_rowpanel<<<64, 32, 0, stream>>>(Mgj, p);
    gj_update<<<64, 256, 0, stream>>>(Mgj, p);
  }
  k_x<<<dim3(4, KB), 256, 0, stream>>>(Mgj, Brho, P4, (float*)d_out);
}